// UniTeacherEncoder_61924838474343
// MI455X (gfx1250) — compile-verified
//
#include <hip/hip_runtime.h>
#include <hip/hip_bf16.h>

typedef __attribute__((ext_vector_type(2))) float v2f;
typedef __attribute__((ext_vector_type(8))) float v8f;

#define GROUPS 8
#define DH 64
#define CG 16
#define NSEQ 1024
#define MSEQ 1024
#define ND 256
#define DIMC 128

__device__ __forceinline__ v8f wmma4(v2f a, v2f b, v8f c) {
  return __builtin_amdgcn_wmma_f32_16x16x4_f32(false, a, false, b, (short)0, c,
                                               false, false);
}

__device__ __forceinline__ float wave_sum(float v) {
  v += __shfl_xor(v, 16, 32);
  v += __shfl_xor(v, 8, 32);
  v += __shfl_xor(v, 4, 32);
  v += __shfl_xor(v, 2, 32);
  v += __shfl_xor(v, 1, 32);
  return v;
}
__device__ __forceinline__ float wave_max(float v) {
  v = fmaxf(v, __shfl_xor(v, 16, 32));
  v = fmaxf(v, __shfl_xor(v, 8, 32));
  v = fmaxf(v, __shfl_xor(v, 4, 32));
  v = fmaxf(v, __shfl_xor(v, 2, 32));
  v = fmaxf(v, __shfl_xor(v, 1, 32));
  return v;
}

// -------------------- LayerNorm rows (dim 128), one wave per row ------------
__global__ void ln_rows_kernel(const float* __restrict__ src,
                               float* __restrict__ dst,
                               const float* __restrict__ g,
                               const float* __restrict__ b, int rows) {
  int lane = threadIdx.x & 31, wid = threadIdx.x >> 5;
  int row = blockIdx.x * 8 + wid;
  if (row >= rows) return;
  const float* p = src + (size_t)row * DIMC;
  float x[4];
#pragma unroll
  for (int i = 0; i < 4; ++i) x[i] = p[lane * 4 + i];
  float s = x[0] + x[1] + x[2] + x[3];
  float m = wave_sum(s) * (1.f / 128.f);
  float vs = 0.f;
#pragma unroll
  for (int i = 0; i < 4; ++i) { float d = x[i] - m; vs += d * d; }
  float var = wave_sum(vs) * (1.f / 128.f);
  float inv = rsqrtf(var + 1e-5f);
#pragma unroll
  for (int i = 0; i < 4; ++i) {
    int c = lane * 4 + i;
    dst[(size_t)row * DIMC + c] = (x[i] - m) * inv * g[c] + b[c];
  }
}

// -------------------- q projection: q[bg][o][n], WMMA ----------------------
__global__ void proj_q_kernel(const float* __restrict__ nx1,
                              const float* __restrict__ qw,
                              float* __restrict__ q) {
  int lane = threadIdx.x & 31, wid = threadIdx.x >> 5;
  int w = blockIdx.x * 8 + wid;               // 16*4*64 = 4096 waves
  int bg = w >> 8, rem = w & 255;
  int ot = rem >> 6, nt = rem & 63;
  int b = bg >> 3, g = bg & 7;
  int hi = lane >> 4, l15 = lane & 15;
  int o = ot * 16 + l15, n = nt * 16 + l15;
  const float* W = qw + g * DH * CG;
  v8f acc = {};
#pragma unroll
  for (int kk = 0; kk < 4; ++kk) {
    int k0 = kk * 4 + hi * 2;
    v2f a, bb;
    a.x = W[o * CG + k0];
    a.y = W[o * CG + k0 + 1];
    const float* src = nx1 + ((size_t)(b * NSEQ + n)) * DIMC + g * CG + k0;
    bb.x = src[0];
    bb.y = src[1];
    acc = wmma4(a, bb, acc);
  }
  size_t base = ((size_t)bg * DH + ot * 16) * NSEQ + nt * 16 + l15;
#pragma unroll
  for (int r = 0; r < 8; ++r)
    q[base + (size_t)(r + hi * 8) * NSEQ] = acc[r];
}

// -------------------- depthwise conv + GELU + tanh offsets ------------------
__global__ void offsets_kernel(const float* __restrict__ q,
                               const float* __restrict__ o1w,
                               const float* __restrict__ o1b,
                               const float* __restrict__ o2w,
                               float* __restrict__ vgs) {
  int lane = threadIdx.x & 31, wid = threadIdx.x >> 5;
  int w = blockIdx.x * 8 + wid;               // 16*256 = 4096 waves
  int bg = w >> 8, nd = w & 255;
  float acc = 0.f;
#pragma unroll
  for (int h = 0; h < 2; ++h) {
    int ch = lane + h * 32;
    const float* gq = q + ((size_t)bg * DH + ch) * NSEQ;
    float s = o1b[ch];
#pragma unroll
    for (int kk = 0; kk < 6; ++kk) {
      int pos = nd * 4 + kk - 1;
      float val = (pos >= 0 && pos < NSEQ) ? gq[pos] : 0.f;
      s += val * o1w[ch * 6 + kk];
    }
    float ge = 0.5f * s * (1.f + erff(s * 0.70710678118654752f));
    acc += ge * o2w[ch];
  }
  acc = wave_sum(acc);
  if (lane == 0) {
    float off = tanhf(acc) * 4.0f;
    float vgrid = (float)nd + off;
    vgs[bg * ND + nd] = 2.f * vgrid / 255.f - 1.f;
  }
}

// -------------------- 1-D bilinear grid sample ------------------------------
__global__ void grid_sample_kernel(const float* __restrict__ nx2,
                                   const float* __restrict__ vgs,
                                   float* __restrict__ kv) {
  int tid = blockIdx.x * 256 + threadIdx.x;   // 16*16*256 = 65536
  int bg = tid >> 12, rem = tid & 4095;
  int c = rem >> 8, j = rem & 255;
  int b = bg >> 3, g = bg & 7;
  float gsv = vgs[bg * ND + j];
  float x = ((gsv + 1.f) * (float)MSEQ - 1.f) * 0.5f;
  float x0 = floorf(x);
  float w1 = x - x0;
  int xi0 = (int)x0, xi1 = xi0 + 1;
  float f0 = (xi0 >= 0 && xi0 <= MSEQ - 1)
                 ? nx2[((size_t)b * MSEQ + xi0) * DIMC + g * CG + c] : 0.f;
  float f1 = (xi1 >= 0 && xi1 <= MSEQ - 1)
                 ? nx2[((size_t)b * MSEQ + xi1) * DIMC + g * CG + c] : 0.f;
  kv[((size_t)bg * CG + c) * ND + j] = f0 * (1.f - w1) + f1 * w1;
}

// -------------------- k/v projection from sampled kv, WMMA ------------------
__global__ void proj_kv_kernel(const float* __restrict__ kv,
                               const float* __restrict__ Wg,
                               float* __restrict__ dst) {
  int lane = threadIdx.x & 31, wid = threadIdx.x >> 5;
  int w = blockIdx.x * 8 + wid;               // 16*4*16 = 1024 waves
  int bg = w >> 6, rem = w & 63;
  int ot = rem >> 4, jt = rem & 15;
  int g = bg & 7;
  int hi = lane >> 4, l15 = lane & 15;
  int o = ot * 16 + l15, j = jt * 16 + l15;
  const float* W = Wg + g * DH * CG;
  v8f acc = {};
#pragma unroll
  for (int kk = 0; kk < 4; ++kk) {
    int k0 = kk * 4 + hi * 2;
    v2f a, bb;
    a.x = W[o * CG + k0];
    a.y = W[o * CG + k0 + 1];
    bb.x = kv[((size_t)bg * CG + k0) * ND + j];
    bb.y = kv[((size_t)bg * CG + k0 + 1) * ND + j];
    acc = wmma4(a, bb, acc);
  }
  size_t base = ((size_t)bg * DH + ot * 16) * ND + jt * 16 + l15;
#pragma unroll
  for (int r = 0; r < 8; ++r) dst[base + (size_t)(r + hi * 8) * ND] = acc[r];
}

// -------------------- sim = (q*scale)^T k, WMMA -----------------------------
__global__ void sim_kernel(const float* __restrict__ q,
                           const float* __restrict__ kmat,
                           float* __restrict__ sim) {
  int lane = threadIdx.x & 31, wid = threadIdx.x >> 5;
  int w = blockIdx.x * 8 + wid;               // 16*64*16 = 16384 waves
  int bg = w >> 10, rem = w & 1023;
  int it = rem >> 4, jt = rem & 15;
  int hi = lane >> 4, l15 = lane & 15;
  int i = it * 16 + l15, j = jt * 16 + l15;
  const float scale = 0.125f;                  // 64^-0.5
  v8f acc = {};
#pragma unroll
  for (int kk = 0; kk < 16; ++kk) {
    int k0 = kk * 4 + hi * 2;
    v2f a, bb;
    a.x = q[((size_t)bg * DH + k0) * NSEQ + i] * scale;
    a.y = q[((size_t)bg * DH + k0 + 1) * NSEQ + i] * scale;
    bb.x = kmat[((size_t)bg * DH + k0) * ND + j];
    bb.y = kmat[((size_t)bg * DH + k0 + 1) * ND + j];
    acc = wmma4(a, bb, acc);
  }
  size_t base = (((size_t)bg << 10) + it * 16) * ND + jt * 16 + l15;
#pragma unroll
  for (int r = 0; r < 8; ++r) sim[base + (size_t)(r + hi * 8) * ND] = acc[r];
}

// -------------------- CPB MLP bias (the FLOP-dominant stage), WMMA ----------
__global__ void cpb_kernel(const float* __restrict__ vgs,
                           const float* __restrict__ w1,
                           const float* __restrict__ b1,
                           const float* __restrict__ W2,
                           const float* __restrict__ b2,
                           const float* __restrict__ w3,
                           const float* __restrict__ b3p,
                           float* __restrict__ sim) {
  __shared__ float h1[8][16][32];
  int lane = threadIdx.x & 31, wid = threadIdx.x >> 5;
  int w = blockIdx.x * 8 + wid;               // 16*1024*16 = 262144 waves
  int jt = w & 15;
  int i = (w >> 4) & 1023;
  int bg = w >> 14;
  int hi = lane >> 4, l15 = lane & 15;
  int j = jt * 16 + l15;
  float gsv = vgs[bg * ND + j];
  float seqi = 2.f * (float)i / 1023.f - 1.f;
  float rel = seqi - gsv;
  float bias = (rel >= 0.f ? 1.f : -1.f) * log1pf(fabsf(rel));
#pragma unroll
  for (int cc = 0; cc < 16; ++cc) {
    int c = hi * 16 + cc;
    h1[wid][l15][c] = fmaxf(bias * w1[c] + b1[c], 0.f);
  }
  __syncthreads();
  v8f acc0, acc1;
  float b2a = b2[l15], b2b = b2[16 + l15];
#pragma unroll
  for (int r = 0; r < 8; ++r) { acc0[r] = b2a; acc1[r] = b2b; }
#pragma unroll
  for (int kk = 0; kk < 8; ++kk) {
    int k0 = kk * 4 + hi * 2;
    v2f a, bb, bc;
    a.x = h1[wid][l15][k0];
    a.y = h1[wid][l15][k0 + 1];
    bb.x = W2[k0 * 32 + l15];
    bb.y = W2[(k0 + 1) * 32 + l15];
    acc0 = wmma4(a, bb, acc0);
    bc.x = W2[k0 * 32 + 16 + l15];
    bc.y = W2[(k0 + 1) * 32 + 16 + l15];
    acc1 = wmma4(a, bc, acc1);
  }
  float w3a = w3[l15], w3b = w3[16 + l15];
  float b3 = b3p[0];
#pragma unroll
  for (int r = 0; r < 8; ++r) {
    float t = fmaxf(acc0[r], 0.f) * w3a + fmaxf(acc1[r], 0.f) * w3b;
    t += __shfl_xor(t, 1, 32);
    t += __shfl_xor(t, 2, 32);
    t += __shfl_xor(t, 4, 32);
    t += __shfl_xor(t, 8, 32);
    if (l15 == 0) {
      int m = r + hi * 8;
      sim[(((size_t)bg << 10) + i) * ND + jt * 16 + m] += t + b3;
    }
  }
}

// -------------------- softmax rows (256 wide), one wave per row -------------
__global__ void softmax_kernel(const float* __restrict__ sim,
                               float* __restrict__ attn) {
  int lane = threadIdx.x & 31, wid = threadIdx.x >> 5;
  int row = blockIdx.x * 8 + wid;              // 16384 rows
  const float* p = sim + (size_t)row * ND;
  float v[8];
  float mx = -1e30f;
#pragma unroll
  for (int i = 0; i < 8; ++i) { v[i] = p[lane + i * 32]; mx = fmaxf(mx, v[i]); }
  mx = wave_max(mx);
  float s = 0.f;
#pragma unroll
  for (int i = 0; i < 8; ++i) { v[i] = expf(v[i] - mx); s += v[i]; }
  s = wave_sum(s);
  float inv = 1.f / s;
#pragma unroll
  for (int i = 0; i < 8; ++i)
    attn[(size_t)row * ND + lane + i * 32] = v[i] * inv;
}

// -------------------- out = attn @ v^T, WMMA --------------------------------
__global__ void attnv_kernel(const float* __restrict__ attn,
                             const float* __restrict__ vmat,
                             float* __restrict__ outg) {
  int lane = threadIdx.x & 31, wid = threadIdx.x >> 5;
  int w = blockIdx.x * 8 + wid;               // 16*64*4 = 4096 waves
  int bg = w >> 8, rem = w & 255;
  int it = rem >> 2, dt = rem & 3;
  int hi = lane >> 4, l15 = lane & 15;
  int i = it * 16 + l15, d = dt * 16 + l15;
  v8f acc = {};
  for (int kk = 0; kk < 64; ++kk) {
    int k0 = kk * 4 + hi * 2;
    v2f a, bb;
    const float* arow = attn + (((size_t)bg << 10) + i) * ND + k0;
    a.x = arow[0];
    a.y = arow[1];
    const float* vrow = vmat + ((size_t)bg * DH + d) * ND + k0;
    bb.x = vrow[0];
    bb.y = vrow[1];
    acc = wmma4(a, bb, acc);
  }
#pragma unroll
  for (int r = 0; r < 8; ++r) {
    int ii = it * 16 + r + hi * 8;
    outg[((size_t)bg * DH + dt * 16 + l15) * NSEQ + ii] = acc[r];
  }
}

// -------------------- xo = ow @ outg + ob, fused residual transpose, WMMA ---
__global__ void outproj_kernel(const float* __restrict__ outg,
                               const float* __restrict__ ow,
                               const float* __restrict__ ob,
                               const float* __restrict__ x1,
                               float* __restrict__ xout) {
  int lane = threadIdx.x & 31, wid = threadIdx.x >> 5;
  int w = blockIdx.x * 8 + wid;               // 2*8*64 = 1024 waves
  int b = w >> 9, rem = w & 511;
  int ot = rem >> 6, nt = rem & 63;
  int hi = lane >> 4, l15 = lane & 15;
  int o = ot * 16 + l15, n = nt * 16 + l15;
  v8f acc = {};
  for (int kk = 0; kk < 128; ++kk) {
    int k0 = kk * 4 + hi * 2;
    v2f a, bb;
    a.x = ow[(size_t)o * 512 + k0];
    a.y = ow[(size_t)o * 512 + k0 + 1];
    bb.x = outg[((size_t)b * 512 + k0) * NSEQ + n];
    bb.y = outg[((size_t)b * 512 + k0 + 1) * NSEQ + n];
    acc = wmma4(a, bb, acc);
  }
#pragma unroll
  for (int r = 0; r < 8; ++r) {
    int oo = ot * 16 + r + hi * 8;
    size_t idx = ((size_t)b * NSEQ + nt * 16 + l15) * DIMC + oo;
    xout[idx] = x1[idx] + acc[r] + ob[oo];
  }
}

// -------------------- fusion: concat(xout0,xout1) @ fus_w^T + fus_b, WMMA ---
__global__ void fusion_kernel(const float* __restrict__ xa,
                              const float* __restrict__ xb,
                              const float* __restrict__ fw,
                              const float* __restrict__ fb,
                              float* __restrict__ xf) {
  int lane = threadIdx.x & 31, wid = threadIdx.x >> 5;
  int w = blockIdx.x * 8 + wid;               // 2*64*8 = 1024 waves
  int b = w >> 9, rem = w & 511;
  int nt = rem >> 3, ot = rem & 7;
  int hi = lane >> 4, l15 = lane & 15;
  int nrow = nt * 16 + l15, o = ot * 16 + l15;
  v8f acc = {};
  for (int kk = 0; kk < 64; ++kk) {
    int k0 = kk * 4 + hi * 2;
    v2f a, bb;
    float a0, a1v;
    if (k0 < 128) {
      const float* r = xa + ((size_t)b * NSEQ + nrow) * DIMC + k0;
      a0 = r[0];
      a1v = (k0 + 1 < 128) ? r[1]
                           : xb[((size_t)b * NSEQ + nrow) * DIMC + 0];
    } else {
      const float* r = xb + ((size_t)b * NSEQ + nrow) * DIMC + (k0 - 128);
      a0 = r[0];
      a1v = r[1];
    }
    a.x = a0;
    a.y = a1v;
    bb.x = fw[(size_t)o * 256 + k0];
    bb.y = fw[(size_t)o * 256 + k0 + 1];
    acc = wmma4(a, bb, acc);
  }
#pragma unroll
  for (int r = 0; r < 8; ++r) {
    int n = nt * 16 + r + hi * 8;
    xf[((size_t)b * NSEQ + n) * DIMC + o] = acc[r] + fb[o];
  }
}

// -------------------- block-wide helpers for 128-thread kernels -------------
__device__ __forceinline__ float block_sum128(float v, float* red) {
  red[threadIdx.x] = v;
  __syncthreads();
  for (int s = 64; s >= 1; s >>= 1) {
    if ((int)threadIdx.x < s) red[threadIdx.x] += red[threadIdx.x + s];
    __syncthreads();
  }
  float r = red[0];
  __syncthreads();
  return r;
}

// -------------------- transfusion v-projection (single-key MHA collapse) ----
__global__ void vproj_kernel(const float* __restrict__ x1a,
                             const float* __restrict__ x1b,
                             const float* __restrict__ ng,
                             const float* __restrict__ nb,
                             const float* __restrict__ tfng,
                             const float* __restrict__ tfnb,
                             const float* __restrict__ inw,
                             const float* __restrict__ inb,
                             const float* __restrict__ oww,
                             const float* __restrict__ obb,
                             float* __restrict__ vproj) {
  __shared__ float s1[128], s2[128], red[128];
  int t = blockIdx.x >> 1, b = blockIdx.x & 1, tid = threadIdx.x;
  const float* row = (t == 0 ? x1a : x1b) + (size_t)b * NSEQ * DIMC;  // n = 0
  float x = row[tid];
  float m = block_sum128(x, red) * (1.f / 128.f);
  float d = x - m;
  float var = block_sum128(d * d, red) * (1.f / 128.f);
  float y1 = d * rsqrtf(var + 1e-5f) * ng[tid] + nb[tid];
  // second LN with tf params
  float m2 = block_sum128(y1, red) * (1.f / 128.f);
  float d2 = y1 - m2;
  float var2 = block_sum128(d2 * d2, red) * (1.f / 128.f);
  s2[tid] = d2 * rsqrtf(var2 + 1e-5f) * tfng[t * 128 + tid] + tfnb[t * 128 + tid];
  __syncthreads();
  const float* Wv = inw + (size_t)t * 384 * 128 + (size_t)(256 + tid) * 128;
  float acc = inb[t * 384 + 256 + tid];
  for (int c = 0; c < 128; ++c) acc += s2[c] * Wv[c];
  s1[tid] = acc;
  __syncthreads();
  const float* Wo = oww + (size_t)t * 128 * 128 + (size_t)tid * 128;
  float acc2 = obb[t * 128 + tid];
  for (int c = 0; c < 128; ++c) acc2 += s1[c] * Wo[c];
  vproj[(size_t)(t * 2 + b) * 128 + tid] = acc2;
}

// -------------------- transfusion LN + mean + head --------------------------
__global__ void tf_reduce_kernel(const float* __restrict__ lnx,
                                 const float* __restrict__ vproj,
                                 const float* __restrict__ tfng,
                                 const float* __restrict__ tfnb,
                                 const float* __restrict__ pw,
                                 const float* __restrict__ pb,
                                 float* __restrict__ out) {
  __shared__ float red[128], sm[128];
  int t = blockIdx.x >> 1, b = blockIdx.x & 1, tid = threadIdx.x;
  float vp = vproj[(size_t)(t * 2 + b) * 128 + tid];
  float g = tfng[t * 128 + tid], be = tfnb[t * 128 + tid];
  float acc = 0.f;
  for (int n = 0; n < NSEQ; ++n) {
    float x = lnx[((size_t)b * NSEQ + n) * DIMC + tid] + vp;
    float m = block_sum128(x, red) * (1.f / 128.f);
    float d = x - m;
    float var = block_sum128(d * d, red) * (1.f / 128.f);
    acc += d * rsqrtf(var + 1e-5f) * g + be;
  }
  sm[tid] = acc * (1.f / 1024.f);
  __syncthreads();
  const float* pwr = pw + (size_t)t * 128 * 128 + (size_t)tid * 128;
  float o = pb[t * 128 + tid];
  for (int c = 0; c < 128; ++c) o += sm[c] * pwr[c];
  out[(size_t)t * 256 + b * 128 + tid] = tanhf(o);
}

// ============================================================================
extern "C" void kernel_launch(void* const* d_in, const int* in_sizes, int n_in,
                              void* d_out, int out_size, void* d_ws,
                              size_t ws_size, hipStream_t stream) {
  const float* x1_0 = (const float*)d_in[0];
  const float* x1_1 = (const float*)d_in[1];
  const float* x2 = (const float*)d_in[2];
  const float* norm_g = (const float*)d_in[3];
  const float* norm_b = (const float*)d_in[4];
  const float* d_qw = (const float*)d_in[5];
  const float* d_kw = (const float*)d_in[6];
  const float* d_vw = (const float*)d_in[7];
  const float* d_ow = (const float*)d_in[8];
  const float* d_ob = (const float*)d_in[9];
  const float* d_o1w = (const float*)d_in[10];
  const float* d_o1b = (const float*)d_in[11];
  const float* d_o2w = (const float*)d_in[12];
  const float* c1w = (const float*)d_in[13];
  const float* c1b = (const float*)d_in[14];
  const float* c2w = (const float*)d_in[15];
  const float* c2b = (const float*)d_in[16];
  const float* c3w = (const float*)d_in[17];
  const float* c3b = (const float*)d_in[18];
  const float* fus_w = (const float*)d_in[19];
  const float* fus_b = (const float*)d_in[20];
  const float* tf_ng = (const float*)d_in[21];
  const float* tf_nb = (const float*)d_in[22];
  const float* tf_inw = (const float*)d_in[23];
  const float* tf_inb = (const float*)d_in[24];
  const float* tf_ow = (const float*)d_in[25];
  const float* tf_ob = (const float*)d_in[26];
  const float* tf_pw = (const float*)d_in[27];
  const float* tf_pb = (const float*)d_in[28];
  (void)in_sizes; (void)n_in; (void)out_size; (void)ws_size;

  float* ws = (float*)d_ws;
  float* nx2 = ws;                       // 2*1024*128
  float* nx1a = nx2 + 262144;            // 2*1024*128
  float* nx1b = nx1a + 262144;
  float* qbuf = nx1b + 262144;           // 16*64*1024
  float* vgs = qbuf + 1048576;           // 16*256
  float* kvbuf = vgs + 4096;             // 16*16*256
  float* kmat = kvbuf + 65536;           // 16*64*256
  float* vmat = kmat + 262144;
  float* simb = vmat + 262144;           // 16*1024*256
  float* outg = simb + 4194304;          // 2*512*1024
  float* xout0 = outg + 1048576;         // 2*1024*128
  float* xout1 = xout0 + 262144;
  float* xfused = xout1 + 262144;
  float* lnx = xfused + 262144;
  float* vproj = lnx + 262144;           // 4*128

  float* fout = (float*)d_out;           // f1(256), f2(256), a1(4194304), a2(...)

  // Stage A: LayerNorms
  ln_rows_kernel<<<256, 256, 0, stream>>>(x2, nx2, norm_g, norm_b, 2048);
  ln_rows_kernel<<<256, 256, 0, stream>>>(x1_0, nx1a, norm_g, norm_b, 2048);
  ln_rows_kernel<<<256, 256, 0, stream>>>(x1_1, nx1b, norm_g, norm_b, 2048);

  for (int t = 0; t < 2; ++t) {
    const float* nx1 = (t == 0) ? nx1a : nx1b;
    const float* x1 = (t == 0) ? x1_0 : x1_1;
    float* xout = (t == 0) ? xout0 : xout1;
    float* attn = fout + 512 + (size_t)t * 4194304;

    proj_q_kernel<<<512, 256, 0, stream>>>(nx1, d_qw + t * 8192, qbuf);
    offsets_kernel<<<512, 256, 0, stream>>>(qbuf, d_o1w + t * 384,
                                            d_o1b + t * 64, d_o2w + t * 64, vgs);
    grid_sample_kernel<<<256, 256, 0, stream>>>(nx2, vgs, kvbuf);
    proj_kv_kernel<<<128, 256, 0, stream>>>(kvbuf, d_kw + t * 8192, kmat);
    proj_kv_kernel<<<128, 256, 0, stream>>>(kvbuf, d_vw + t * 8192, vmat);
    sim_kernel<<<2048, 256, 0, stream>>>(qbuf, kmat, simb);
    cpb_kernel<<<32768, 256, 0, stream>>>(vgs, c1w + t * 32, c1b + t * 32,
                                          c2w + t * 1024, c2b + t * 32,
                                          c3w + t * 32, c3b + t, simb);
    softmax_kernel<<<2048, 256, 0, stream>>>(simb, attn);
    attnv_kernel<<<512, 256, 0, stream>>>(attn, vmat, outg);
    outproj_kernel<<<128, 256, 0, stream>>>(outg, d_ow + t * 65536,
                                            d_ob + t * 128, x1, xout);
  }

  fusion_kernel<<<128, 256, 0, stream>>>(xout0, xout1, fus_w, fus_b, xfused);
  ln_rows_kernel<<<256, 256, 0, stream>>>(xfused, lnx, norm_g, norm_b, 2048);
  vproj_kernel<<<4, 128, 0, stream>>>(x1_0, x1_1, norm_g, norm_b, tf_ng, tf_nb,
                                      tf_inw, tf_inb, tf_ow, tf_ob, vproj);
  tf_reduce_kernel<<<4, 128, 0, stream>>>(lnx, vproj, tf_ng, tf_nb, tf_pw,
                                          tf_pb, fout);
}